// MulticlassTargetEncoder_17489106830034
// MI455X (gfx1250) — compile-verified
//
#include <hip/hip_runtime.h>
#include <hip/hip_bf16.h>
#include <stdint.h>

// Problem constants (match reference: B=128, S=8192, F=32, N_CLASSES=16)
#define BB 128
#define SS 8192
#define FF 32
#define ELEMS_PER_BATCH (SS * FF)          // 262144 floats per batch row

// Native clang vector type: accepted by __builtin_nontemporal_store
// (HIP's float4 struct is not). Same 16-byte layout as float4.
typedef float v4f __attribute__((ext_vector_type(4)));

// ---------------- Phase 1: per-batch 16-bit class-presence mask ----------------
#define P1_BLOCKS_PER_BATCH 16
#define P1_THREADS 256

__global__ void presence_kernel(const float* __restrict__ x,
                                const int* __restrict__ eval_pos_p,
                                unsigned* __restrict__ masks) {
    const int b     = blockIdx.x / P1_BLOCKS_PER_BATCH;
    const int chunk = blockIdx.x % P1_BLOCKS_PER_BATCH;
    const int trainElems = eval_pos_p[0] * FF;   // contiguous prefix of the batch row

    const float* __restrict__ xb = x + (size_t)b * ELEMS_PER_BATCH;

    unsigned m = 0u;
    const int stride = P1_BLOCKS_PER_BATCH * P1_THREADS * 4;
    for (int i = (chunk * P1_THREADS + (int)threadIdx.x) * 4; i + 3 < trainElems; i += stride) {
        v4f v = *(const v4f*)(xb + i);
        m |= 1u << (unsigned)v.x;
        m |= 1u << (unsigned)v.y;
        m |= 1u << (unsigned)v.z;
        m |= 1u << (unsigned)v.w;
    }

    __shared__ unsigned sm;
    if (threadIdx.x == 0) sm = 0u;
    __syncthreads();
    atomicOr(&sm, m);
    __syncthreads();
    if (threadIdx.x == 0) atomicOr(&masks[b], sm);
}

// ---------------- Phase 2: rank every element via popcount ----------------
// 4-deep async global->LDS ring buffer (CDNA5 async data mover, ASYNCcnt).
#define P2_THREADS 256
#define P2_STAGES  16
#define P2_DEPTH   4
#define P2_CHUNK   (P2_THREADS * 4 * P2_STAGES)            // 16384 elems per block
#define P2_BLOCKS_PER_BATCH (ELEMS_PER_BATCH / P2_CHUNK)   // 16

__device__ __forceinline__ void async_copy16(unsigned ldsAddr, const v4f* g) {
    // Copies 16 bytes from global address g to the LDS address in ldsAddr, tracked with ASYNCcnt.
    asm volatile("global_load_async_to_lds_b128 %0, %1, off"
                 :: "v"(ldsAddr), "v"(g)
                 : "memory");
}

__device__ __forceinline__ void wait_async(int n) {
    // n is a compile-time constant under full unroll; branches fold away.
    if      (n <= 0) asm volatile("s_wait_asynccnt 0x0" ::: "memory");
    else if (n == 1) asm volatile("s_wait_asynccnt 0x1" ::: "memory");
    else if (n == 2) asm volatile("s_wait_asynccnt 0x2" ::: "memory");
    else             asm volatile("s_wait_asynccnt 0x3" ::: "memory");
}

__global__ void rank_kernel(const float* __restrict__ x,
                            const unsigned* __restrict__ masks,
                            float* __restrict__ out) {
    const int b     = blockIdx.x / P2_BLOCKS_PER_BATCH;
    const int chunk = blockIdx.x % P2_BLOCKS_PER_BATCH;
    const unsigned m = masks[b];                 // uniform -> scalar load

    const size_t base = (size_t)b * ELEMS_PER_BATCH + (size_t)chunk * P2_CHUNK;
    const v4f* __restrict__ gin  = (const v4f*)(x + base);
    v4f* __restrict__       gout = (v4f*)(out + base);

    __shared__ v4f buf[P2_DEPTH][P2_THREADS];   // 16 KB ring
    const int t = threadIdx.x;
    unsigned ldsAddr[P2_DEPTH];
#pragma unroll
    for (int i = 0; i < P2_DEPTH; ++i)
        ldsAddr[i] = (unsigned)(uintptr_t)&buf[i][t];

    // Prime the pipeline: stages 0 .. DEPTH-2 in flight (3 outstanding).
#pragma unroll
    for (int i = 0; i < P2_DEPTH - 1; ++i)
        async_copy16(ldsAddr[i], gin + i * P2_THREADS + t);

#pragma unroll
    for (int s = 0; s < P2_STAGES; ++s) {
        if (s + P2_DEPTH - 1 < P2_STAGES) {
            // Issue stage s+3 into the ring (up to 4 outstanding), then wait
            // until at most 3 remain -> oldest (stage s) has landed.
            async_copy16(ldsAddr[(s + P2_DEPTH - 1) % P2_DEPTH],
                         gin + (s + P2_DEPTH - 1) * P2_THREADS + t);
            wait_async(P2_DEPTH - 1);
        } else {
            // Drain: stages s..STAGES-1 outstanding; need stage s done.
            wait_async(P2_STAGES - 1 - s);
        }

        // Each wave reads only LDS slots it staged itself -> no barrier needed.
        v4f v = buf[s % P2_DEPTH][t];
        v4f r;
        r.x = (float)__popc(m & ((1u << (unsigned)v.x) - 1u));
        r.y = (float)__popc(m & ((1u << (unsigned)v.y) - 1u));
        r.z = (float)__popc(m & ((1u << (unsigned)v.z) - 1u));
        r.w = (float)__popc(m & ((1u << (unsigned)v.w) - 1u));

        // Output is write-once: non-temporal store keeps the 64MB training
        // prefix resident in the 192MB L2 between phase 1 and phase 2 reads.
        __builtin_nontemporal_store(r, gout + s * P2_THREADS + t);
    }
}

// ---------------- Launch ----------------
extern "C" void kernel_launch(void* const* d_in, const int* in_sizes, int n_in,
                              void* d_out, int out_size, void* d_ws, size_t ws_size,
                              hipStream_t stream) {
    (void)in_sizes; (void)n_in; (void)out_size; (void)ws_size;

    const float* x        = (const float*)d_in[0];
    const int*   eval_pos = (const int*)d_in[1];
    float*       out      = (float*)d_out;
    unsigned*    masks    = (unsigned*)d_ws;     // 128 * 4 bytes

    // Zero the presence masks each call (capture-legal, replayed every launch).
    (void)hipMemsetAsync(masks, 0, BB * sizeof(unsigned), stream);

    presence_kernel<<<BB * P1_BLOCKS_PER_BATCH, P1_THREADS, 0, stream>>>(x, eval_pos, masks);
    rank_kernel<<<BB * P2_BLOCKS_PER_BATCH, P2_THREADS, 0, stream>>>(x, masks, out);
}